// SoftKNN_9904194584583
// MI455X (gfx1250) — compile-verified
//
#include <hip/hip_runtime.h>
#include <math.h>

// ---------------------------------------------------------------------------
// Fused soft-KNN for MI455X (gfx1250, wave32).
//   Phase 0: ynorm[n] = sum_k train[n][k]^2                        (d_ws)
//   Phase 1: grid (nq/16, NSEG). Each workgroup: one 16-query tile x one
//            N-segment. Stream train tiles through V_WMMA_F32_16X16X4_F32,
//            keep running top-5 of (||y||^2 - 2 x.y) per row in registers,
//            merge waves via LDS, write per-segment top-5 to d_ws.
//   Phase 2: one lane per query row: merge NSEG*5 candidates, add ||x||^2,
//            sqrt, softmax(-d/T), label vote.
// ---------------------------------------------------------------------------

typedef __attribute__((ext_vector_type(2))) float v2f;
typedef __attribute__((ext_vector_type(8))) float v8f;

#define DIMS    128
#define KSTEPS  (DIMS / 4)      // 32 wmma_f32_16x16x4 per 16x16 tile
#define KNN     5
#define NCLS    10
#define NW      4               // waves per workgroup
#define NSEG    4               // N-range segments (occupancy)
#define WSTRIDE (NW * NSEG)     // tile stride per worker wave
#define TEMP    1.0f
#define FLT_BIG 3.4e38f

// workspace layout (bytes, 256-aligned)
#define WS_YNORM_OFF 0
#define WS_KEY_OFF   200192                  // >= 50000*4, 256-aligned
#define WS_IDX_OFF   (WS_KEY_OFF + 327680)   // >= NSEG*KNN*4096*4

// ---------------- Phase 0: train-row squared norms -------------------------
__global__ void ynorm_kernel(const float* __restrict__ train,
                             float* __restrict__ ynorm, int n) {
    int i = blockIdx.x * blockDim.x + threadIdx.x;
    if (i >= n) return;
    const float4* p = (const float4*)(train + (size_t)i * DIMS);
    float s = 0.f;
#pragma unroll
    for (int j = 0; j < DIMS / 4; ++j) {
        float4 v = p[j];
        s += v.x * v.x + v.y * v.y + v.z * v.z + v.w * v.w;
    }
    ynorm[i] = s;
}

// ---------------- Phase 1: fused GEMM + per-segment top-k ------------------
__global__ __launch_bounds__(NW * 32)
void knn_partial_kernel(const float* __restrict__ x,
                        const float* __restrict__ train,
                        const float* __restrict__ ynorm,
                        float* __restrict__ ws_key,
                        int*   __restrict__ ws_idx,
                        int n_train, int nq) {
    __shared__ float s_key[16][NW * 16 * KNN];   // 16 rows x 320 candidates
    __shared__ int   s_id [16][NW * 16 * KNN];

    const int tid  = threadIdx.x;
    const int wave = tid >> 5;       // wave32
    const int lane = tid & 31;
    const int lg   = lane >> 4;      // lane group: K-pair select (A/B), +8 rows (C)
    const int ln   = lane & 15;      // matrix row (A) / column (B,C)
    const int row_base = blockIdx.x * 16;
    const int seg      = blockIdx.y;
    const int worker   = seg * NW + wave;        // 0..WSTRIDE-1

    // ---- A fragments: lane holds A[ln][4k + 2*lg + {0,1}], k = 0..31 ----
    v2f a[KSTEPS];
    const float* xrow = x + (size_t)(row_base + ln) * DIMS;
#pragma unroll
    for (int k = 0; k < KSTEPS; ++k)
        a[k] = *(const v2f*)(xrow + 4 * k + 2 * lg);

    // ---- running top-KNN of key = ||y||^2 - 2 x.y  (||x||^2 added later) ----
    float td[8][KNN];
    int   ti[8][KNN];
#pragma unroll
    for (int r = 0; r < 8; ++r)
#pragma unroll
        for (int j = 0; j < KNN; ++j) { td[r][j] = FLT_BIG; ti[r][j] = 0; }

    const int ntiles = n_train / 16;
    for (int t = worker; t < ntiles; t += WSTRIDE) {
        const int col0 = t * 16;
        const float* brow = train + (size_t)(col0 + ln) * DIMS;

        // pull this worker's NEXT 8KB B tile into the WGP cache:
        // lanes spread 256B apart (+128B second issue) to cover all lines.
        if (t + WSTRIDE < ntiles) {
            const char* pf = (const char*)(train + (size_t)(col0 + WSTRIDE * 16) * DIMS);
            __builtin_prefetch(pf + lane * 256, 0, 3);
            __builtin_prefetch(pf + lane * 256 + 128, 0, 3);
        }

        v8f c = {0.f, 0.f, 0.f, 0.f, 0.f, 0.f, 0.f, 0.f};
#pragma unroll
        for (int k = 0; k < KSTEPS; ++k) {
            // B[K][N]: lane (lg*16+ln) holds B[4k + 2*lg + {0,1}][ln]
            v2f b = *(const v2f*)(brow + 4 * k + 2 * lg);
            c = __builtin_amdgcn_wmma_f32_16x16x4_f32(
                    false, a[k], false, b, (short)0, c, false, false);
        }

        const float yn   = ynorm[col0 + ln];
        const int   colg = col0 + ln;
#pragma unroll
        for (int r = 0; r < 8; ++r) {
            float d  = fmaf(-2.f, c[r], yn);     // ordering key
            int   id = colg;
#pragma unroll
            for (int j = 0; j < KNN; ++j) {      // branchless sorted insert
                float cur = td[r][j]; int ci = ti[r][j];
                bool  lt  = d < cur;
                td[r][j] = lt ? d  : cur;
                ti[r][j] = lt ? id : ci;
                d  = lt ? cur : d;
                id = lt ? ci  : id;
            }
        }
    }

    // ---- dump per-lane partial lists to LDS ----
#pragma unroll
    for (int r = 0; r < 8; ++r) {
        const int row  = r + 8 * lg;
        const int base = (wave * 16 + ln) * KNN;
#pragma unroll
        for (int j = 0; j < KNN; ++j) {
            s_key[row][base + j] = td[r][j];
            s_id [row][base + j] = ti[r][j];
        }
    }
    __syncthreads();

    // ---- per-row merge of NW*16 partial lists -> segment top-5 ----
    if (tid < 16) {
        const int row = tid;
        float bd[KNN]; int bi[KNN];
#pragma unroll
        for (int j = 0; j < KNN; ++j) { bd[j] = FLT_BIG; bi[j] = 0; }
        for (int e = 0; e < NW * 16 * KNN; ++e) {
            float d  = s_key[row][e];
            int   id = s_id [row][e];
#pragma unroll
            for (int j = 0; j < KNN; ++j) {
                float cur = bd[j]; int ci = bi[j];
                bool  lt  = d < cur;
                bd[j] = lt ? d  : cur;
                bi[j] = lt ? id : ci;
                d  = lt ? cur : d;
                id = lt ? ci  : id;
            }
        }
        // layout [seg*KNN + j][nq] so the final kernel reads coalesced
#pragma unroll
        for (int j = 0; j < KNN; ++j) {
            const size_t slot = (size_t)(seg * KNN + j) * nq + row_base + row;
            ws_key[slot] = bd[j];
            ws_idx[slot] = bi[j];
        }
    }
}

// ---------------- Phase 2: cross-segment merge + softmax + vote ------------
__global__ void knn_final_kernel(const float* __restrict__ x,
                                 const int*   __restrict__ labels,
                                 const float* __restrict__ ws_key,
                                 const int*   __restrict__ ws_idx,
                                 float* __restrict__ out, int nq) {
    const int row = blockIdx.x * blockDim.x + threadIdx.x;
    if (row >= nq) return;

    // ||x_row||^2
    const float4* p = (const float4*)(x + (size_t)row * DIMS);
    float xn = 0.f;
#pragma unroll
    for (int j = 0; j < DIMS / 4; ++j) {
        float4 v = p[j];
        xn += v.x * v.x + v.y * v.y + v.z * v.z + v.w * v.w;
    }

    // merge NSEG*KNN candidates (coalesced reads)
    float bd[KNN]; int bi[KNN];
#pragma unroll
    for (int j = 0; j < KNN; ++j) { bd[j] = FLT_BIG; bi[j] = 0; }
#pragma unroll
    for (int e = 0; e < NSEG * KNN; ++e) {
        float d  = ws_key[(size_t)e * nq + row];
        int   id = ws_idx[(size_t)e * nq + row];
#pragma unroll
        for (int j = 0; j < KNN; ++j) {
            float cur = bd[j]; int ci = bi[j];
            bool  lt  = d < cur;
            bd[j] = lt ? d  : cur;
            bi[j] = lt ? id : ci;
            d  = lt ? cur : d;
            id = lt ? ci  : id;
        }
    }

    // dist = sqrt(max(xn + key, 0)); softmax(-dist/T); weighted one-hot vote
    float dist[KNN], w[KNN], sum = 0.f;
#pragma unroll
    for (int j = 0; j < KNN; ++j) dist[j] = sqrtf(fmaxf(xn + bd[j], 0.f));
    const float dmin = dist[0];              // list sorted ascending
#pragma unroll
    for (int j = 0; j < KNN; ++j) { w[j] = expf((dmin - dist[j]) / TEMP); sum += w[j]; }
    const float inv = 1.f / sum;

    int lab[KNN];
#pragma unroll
    for (int j = 0; j < KNN; ++j) lab[j] = labels[bi[j]];

    float* o = out + (size_t)row * NCLS;
#pragma unroll
    for (int cc = 0; cc < NCLS; ++cc) {
        float acc = 0.f;
#pragma unroll
        for (int j = 0; j < KNN; ++j) acc += (lab[j] == cc) ? w[j] * inv : 0.f;
        o[cc] = acc;
    }
}

// ---------------------------------------------------------------------------
extern "C" void kernel_launch(void* const* d_in, const int* in_sizes, int n_in,
                              void* d_out, int out_size, void* d_ws, size_t ws_size,
                              hipStream_t stream) {
    const float* x      = (const float*)d_in[0];   // [4096, 128]
    const float* train  = (const float*)d_in[1];   // [50000, 128]
    const int*   labels = (const int*)d_in[2];     // [50000]
    float*       out    = (float*)d_out;           // [4096, 10]

    char* ws = (char*)d_ws;
    float* ynorm  = (float*)(ws + WS_YNORM_OFF);   // [n_train]
    float* ws_key = (float*)(ws + WS_KEY_OFF);     // [NSEG*KNN, nq]
    int*   ws_idx = (int*)  (ws + WS_IDX_OFF);     // [NSEG*KNN, nq]

    const int n_train = in_sizes[2];
    const int nq      = in_sizes[0] / DIMS;

    ynorm_kernel<<<(n_train + 255) / 256, 256, 0, stream>>>(train, ynorm, n_train);
    knn_partial_kernel<<<dim3(nq / 16, NSEG), NW * 32, 0, stream>>>(
        x, train, ynorm, ws_key, ws_idx, n_train, nq);
    knn_final_kernel<<<(nq + 255) / 256, 256, 0, stream>>>(
        x, labels, ws_key, ws_idx, out, nq);
}